// CeptaBlock_26843545600399
// MI455X (gfx1250) — compile-verified
//
#include <hip/hip_runtime.h>

#define B_ 4
#define T_ 2048
#define D_ 2048
#define P_ 512
#define K_ 16
#define ALPHA_ 4
#define R_ 32
#define NE_ 409              // int(0.8 * 512)
#define ROWS_ (B_*T_)        // 8192
#define NC_ 64               // scan chunks
#define LCH_ (T_/NC_)        // 32 steps per chunk
#define EPS_ 1e-6f

typedef __attribute__((ext_vector_type(2))) float v2f;
typedef __attribute__((ext_vector_type(4))) float v4f;
typedef __attribute__((ext_vector_type(8))) float v8f;

// fast sigmoid: v_exp_f32 + v_rcp_f32 (avoids IEEE div-fixup sequence)
__device__ __forceinline__ float sigmoidf_(float x) {
    return __builtin_amdgcn_rcpf(1.0f + __expf(-x));
}

// ---------------------------------------------------------------------------
// K0: per-neuron quant_dale scalars: c = sign*mean|Wp|, d = sign*mean|Wout|
// ---------------------------------------------------------------------------
__global__ void k0_precompute(const float* __restrict__ Wp_ssm, const float* __restrict__ Wout_ssm,
                              const float* __restrict__ Wp_mlp, const float* __restrict__ Wout_mlp,
                              float* __restrict__ cd) {
    int p = threadIdx.x;
    if (p >= P_) return;
    float sgn = (p < NE_) ? 1.0f : -1.0f;
    float s;
    s = 0.f; for (int k = 0; k < K_; k++)     s += fabsf(Wp_ssm[p*K_+k]);       cd[0*P_+p] = sgn * s * (1.0f/K_);
    s = 0.f; for (int a = 0; a < ALPHA_; a++) s += fabsf(Wout_ssm[p*ALPHA_+a]); cd[1*P_+p] = sgn * s * (1.0f/ALPHA_);
    s = 0.f; for (int k = 0; k < K_; k++)     s += fabsf(Wp_mlp[p*K_+k]);       cd[2*P_+p] = sgn * s * (1.0f/K_);
    s = 0.f; for (int a = 0; a < ALPHA_; a++) s += fabsf(Wout_mlp[p*ALPHA_+a]); cd[3*P_+p] = sgn * s * (1.0f/ALPHA_);
}

// ---------------------------------------------------------------------------
// K1: fused rmsnorm(x,g1) + synapse gather + sigmoid gate (SSM branch)
// one block per (b,t) row; h1 lives only in LDS. b128 row traffic.
// ---------------------------------------------------------------------------
__global__ __launch_bounds__(256) void k1_norm_gather_ssm(
    const float* __restrict__ x, const float* __restrict__ g1,
    const int* __restrict__ idx, const float* __restrict__ wg,
    const float* __restrict__ cd,
    float* __restrict__ f_out, float* __restrict__ t_out) {
    __shared__ float h[D_];
    __shared__ float sred[9];
    int row = blockIdx.x, tid = threadIdx.x;
    const v4f* xr4 = (const v4f*)(x + (size_t)row * D_);
    float ss = 0.f;
    #pragma unroll
    for (int i = tid; i < D_/4; i += 256) {
        v4f v = xr4[i];
        *(v4f*)&h[4*i] = v;
        ss += v.x*v.x + v.y*v.y + v.z*v.z + v.w*v.w;
    }
    for (int off = 16; off > 0; off >>= 1) ss += __shfl_xor(ss, off, 32);
    if ((tid & 31) == 0) sred[tid >> 5] = ss;
    __syncthreads();
    if (tid == 0) { float t = 0.f; for (int i = 0; i < 8; i++) t += sred[i]; sred[8] = t; }
    __syncthreads();
    float rn = rsqrtf(sred[8] * (1.0f / D_) + EPS_);
    const v4f* g14 = (const v4f*)g1;
    #pragma unroll
    for (int i = tid; i < D_/4; i += 256) {
        v4f v = *(v4f*)&h[4*i];
        v4f g = g14[i];
        v.x *= rn*g.x; v.y *= rn*g.y; v.z *= rn*g.z; v.w *= rn*g.w;
        *(v4f*)&h[4*i] = v;
    }
    __syncthreads();
    #pragma unroll
    for (int pp = 0; pp < 2; pp++) {
        int p = tid * 2 + pp;
        const int4* ip = (const int4*)(idx + p*K_);   // 64B aligned (K=16)
        const v4f*  wp = (const v4f*) (wg  + p*K_);
        float u = 0.f;
        #pragma unroll
        for (int q = 0; q < K_/4; q++) {
            int4 ii = ip[q]; v4f ww = wp[q];
            u += h[ii.x]*ww.x + h[ii.y]*ww.y + h[ii.z]*ww.z + h[ii.w]*ww.w;
        }
        u *= cd[p];
        float f = sigmoidf_(u);
        f_out[(size_t)row*P_ + p] = f;
        t_out[(size_t)row*P_ + p] = f * u;
    }
}

// ---------------------------------------------------------------------------
// K2: z = t_ssm @ U ; lam = sigmoid(f_ssm @ Wgate + bgate)   [8192,512]x[512,32]
// 4 waves/block: waves 0,1 -> z N-tiles, waves 2,3 -> lam N-tiles. WMMA f32 16x16x4.
// Epilogue is branch-free (select) so EXEC stays full throughout.
// ---------------------------------------------------------------------------
__global__ __launch_bounds__(128) void k2_wmma_zlam(
    const float* __restrict__ t_in, const float* __restrict__ f_in,
    const float* __restrict__ U, const float* __restrict__ Wg,
    const float* __restrict__ bgate,
    float* __restrict__ z, float* __restrict__ lam) {
    int tid = threadIdx.x;
    int w = tid >> 5, lane = tid & 31;
    int mt = blockIdx.x;
    int nt = w & 1;
    bool is_lam = (w >= 2);
    const float* A   = is_lam ? f_in : t_in;
    const float* Bm  = is_lam ? Wg   : U;
    float*       dst = is_lam ? lam  : z;
    int arow = mt * 16 + (lane & 15);
    int koff = (lane >> 4) * 2;
    int n = nt * 16 + (lane & 15);          // global r index
    const float* Arow = A + (size_t)arow * P_;
    v8f acc = {};
    for (int k0 = 0; k0 < P_; k0 += 4) {
        int kb = k0 + koff;
        v2f a = *(const v2f*)(Arow + kb);
        v2f b; b.x = Bm[(size_t)kb * R_ + n]; b.y = Bm[(size_t)(kb + 1) * R_ + n];
        acc = __builtin_amdgcn_wmma_f32_16x16x4_f32(false, a, false, b, (short)0, acc, false, false);
    }
    float bg = bgate[n];
    #pragma unroll
    for (int j = 0; j < 8; j++) {
        int row = mt * 16 + j + ((lane >= 16) ? 8 : 0);
        size_t o = (size_t)row * R_ + n;
        float raw = acc[j];
        float sv  = sigmoidf_(raw + bg);          // computed unconditionally: no exec-mask dance
        dst[o] = is_lam ? sv : raw;
    }
}

// ---------------------------------------------------------------------------
// K3: chunked parallel linear scan  s = lam*s + z  over T (B*R chains)
// ---------------------------------------------------------------------------
__global__ void k3a_scan_local(const float* __restrict__ z, const float* __restrict__ lam,
                               float* __restrict__ Sloc, float* __restrict__ Pp) {
    int b = blockIdx.x / NC_, c = blockIdx.x % NC_;
    int r = threadIdx.x;                    // 32 threads = R lanes
    float s = 0.f, pl = 1.f;
    size_t base = ((size_t)b * T_ + (size_t)c * LCH_) * R_ + r;
    for (int i = 0; i < LCH_; i++) {
        size_t o = base + (size_t)i * R_;
        float lv = lam[o];
        s = lv * s + z[o];
        pl *= lv;
        Sloc[o] = s; Pp[o] = pl;
    }
}

__global__ void k3b_scan_chunks(const float* __restrict__ Sloc, const float* __restrict__ Pp,
                                const float* __restrict__ state, float* __restrict__ carry,
                                float* __restrict__ out_state) {
    int tid = threadIdx.x;                  // 128 = B*R
    int b = tid >> 5, r = tid & 31;
    float s = state[b * R_ + r];
    for (int c = 0; c < NC_; c++) {
        carry[((size_t)b * NC_ + c) * R_ + r] = s;
        size_t oe = ((size_t)b * T_ + (size_t)c * LCH_ + (LCH_ - 1)) * R_ + r;
        s = Pp[oe] * s + Sloc[oe];
    }
    out_state[b * R_ + r] = s;              // new_state -> d_out tail
}

__global__ __launch_bounds__(256) void k3c_scan_fix(float* __restrict__ S, const float* __restrict__ Pp,
                                                    const float* __restrict__ carry) {
    size_t i = (size_t)blockIdx.x * 256 + threadIdx.x;   // over B*T*R
    int r = (int)(i % R_);
    size_t bt = i / R_;
    int t = (int)(bt % T_);
    int b = (int)(bt / T_);
    int c = t / LCH_;
    S[i] = S[i] + carry[((size_t)b * NC_ + c) * R_ + r] * Pp[i];
}

// ---------------------------------------------------------------------------
// K4: tilde = S @ V  [8192,32]x[32,512]  WMMA, fused epilogue:
//     x_after[row, 4p+a] = x[row, 4p+a] + f*tilde*d_p   -> written to d_out
//     (x_after stays RT-cached: 67MB fits the 192MB L2 and k5 re-reads it)
// ---------------------------------------------------------------------------
__global__ __launch_bounds__(128) void k4_wmma_tilde_residual(
    const float* __restrict__ S, const float* __restrict__ V,
    const float* __restrict__ f_in, const float* __restrict__ d_ssm,
    const float* __restrict__ x, float* __restrict__ xout) {
    int tid = threadIdx.x;
    int w = tid >> 5, lane = tid & 31;
    int mt = blockIdx.x;
    int arow = mt * 16 + (lane & 15);
    int koff = (lane >> 4) * 2;
    const float* Sr = S + (size_t)arow * R_;
    v2f afr[8];
    #pragma unroll
    for (int kk = 0; kk < 8; kk++) afr[kk] = *(const v2f*)(Sr + kk * 4 + koff);
    for (int nt = w; nt < 32; nt += 4) {    // uniform per wave -> EXEC stays full
        int n = nt * 16 + (lane & 15);      // global p
        v8f acc = {};
        #pragma unroll
        for (int kk = 0; kk < 8; kk++) {
            int kb = kk * 4 + koff;
            v2f b; b.x = V[(size_t)kb * P_ + n]; b.y = V[(size_t)(kb + 1) * P_ + n];
            acc = __builtin_amdgcn_wmma_f32_16x16x4_f32(false, afr[kk], false, b, (short)0, acc, false, false);
        }
        float dv = d_ssm[n];
        #pragma unroll
        for (int j = 0; j < 8; j++) {
            int row = mt * 16 + j + ((lane >= 16) ? 8 : 0);
            float fv = f_in[(size_t)row * P_ + n];
            float val = fv * acc[j] * dv;
            v4f xv = *(const v4f*)(x + (size_t)row * D_ + 4 * n);
            v4f ov; ov.x = xv.x + val; ov.y = xv.y + val; ov.z = xv.z + val; ov.w = xv.w + val;
            *(v4f*)(xout + (size_t)row * D_ + 4 * n) = ov;
        }
    }
}

// ---------------------------------------------------------------------------
// K5: fused MLP branch: rmsnorm(x_after,g2) + gather + gate + residual,
// in place on d_out (row buffered in LDS before any write -> race-free).
// Final stores are nontemporal: x_out is never re-read, keep L2 for x_after.
// ---------------------------------------------------------------------------
__global__ __launch_bounds__(256) void k5_mlp(
    float* __restrict__ xio, const float* __restrict__ g2,
    const int* __restrict__ idx, const float* __restrict__ wg,
    const float* __restrict__ c_mlp, const float* __restrict__ d_mlp) {
    __shared__ float xa[D_];
    __shared__ float h[D_];
    __shared__ float sred[9];
    int row = blockIdx.x, tid = threadIdx.x;
    float* xr = xio + (size_t)row * D_;
    const v4f* xr4 = (const v4f*)xr;
    float ss = 0.f;
    #pragma unroll
    for (int i = tid; i < D_/4; i += 256) {
        v4f v = xr4[i];
        *(v4f*)&xa[4*i] = v;
        ss += v.x*v.x + v.y*v.y + v.z*v.z + v.w*v.w;
    }
    for (int off = 16; off > 0; off >>= 1) ss += __shfl_xor(ss, off, 32);
    if ((tid & 31) == 0) sred[tid >> 5] = ss;
    __syncthreads();
    if (tid == 0) { float t = 0.f; for (int i = 0; i < 8; i++) t += sred[i]; sred[8] = t; }
    __syncthreads();
    float rn = rsqrtf(sred[8] * (1.0f / D_) + EPS_);
    const v4f* g24 = (const v4f*)g2;
    #pragma unroll
    for (int i = tid; i < D_/4; i += 256) {
        v4f v = *(v4f*)&xa[4*i];
        v4f g = g24[i];
        v.x *= rn*g.x; v.y *= rn*g.y; v.z *= rn*g.z; v.w *= rn*g.w;
        *(v4f*)&h[4*i] = v;
    }
    __syncthreads();
    #pragma unroll
    for (int pp = 0; pp < 2; pp++) {
        int p = tid * 2 + pp;
        const int4* ip = (const int4*)(idx + p*K_);
        const v4f*  wp = (const v4f*) (wg  + p*K_);
        float u = 0.f;
        #pragma unroll
        for (int q = 0; q < K_/4; q++) {
            int4 ii = ip[q]; v4f ww = wp[q];
            u += h[ii.x]*ww.x + h[ii.y]*ww.y + h[ii.z]*ww.z + h[ii.w]*ww.w;
        }
        u *= c_mlp[p];
        float f = sigmoidf_(u);
        float y = f * u * d_mlp[p];
        v4f ov;
        ov.x = xa[4*p+0] + y; ov.y = xa[4*p+1] + y; ov.z = xa[4*p+2] + y; ov.w = xa[4*p+3] + y;
        __builtin_nontemporal_store(ov, (v4f*)(xr + 4 * p));
    }
}

// ---------------------------------------------------------------------------
extern "C" void kernel_launch(void* const* d_in, const int* in_sizes, int n_in,
                              void* d_out, int out_size, void* d_ws, size_t ws_size,
                              hipStream_t stream) {
    const float* x        = (const float*)d_in[0];
    const float* state    = (const float*)d_in[1];
    const float* g1       = (const float*)d_in[2];
    const float* g2       = (const float*)d_in[3];
    const int*   idx_ssm  = (const int*)  d_in[4];
    const float* wg_ssm   = (const float*)d_in[5];
    const int*   idx_mlp  = (const int*)  d_in[6];
    const float* wg_mlp   = (const float*)d_in[7];
    const float* Wp_ssm   = (const float*)d_in[8];
    const float* Wout_ssm = (const float*)d_in[9];
    const float* Wp_mlp   = (const float*)d_in[10];
    const float* Wout_mlp = (const float*)d_in[11];
    const float* U        = (const float*)d_in[12];
    const float* V        = (const float*)d_in[13];
    const float* Wgate    = (const float*)d_in[14];
    const float* bgate    = (const float*)d_in[15];

    float* out = (float*)d_out;            // [B*T*D] x_out, then [B*R] new_state
    float* ws  = (float*)d_ws;

    size_t o = 0;
    float* cd    = ws + o; o += 4 * P_;
    float* f_ssm = ws + o; o += (size_t)ROWS_ * P_;
    float* t_ssm = ws + o; o += (size_t)ROWS_ * P_;
    float* z     = ws + o; o += (size_t)ROWS_ * R_;
    float* lam   = ws + o; o += (size_t)ROWS_ * R_;
    float* Sloc  = ws + o; o += (size_t)ROWS_ * R_;
    float* Pp    = ws + o; o += (size_t)ROWS_ * R_;
    float* carry = ws + o; o += (size_t)B_ * NC_ * R_;

    k0_precompute<<<1, P_, 0, stream>>>(Wp_ssm, Wout_ssm, Wp_mlp, Wout_mlp, cd);
    k1_norm_gather_ssm<<<ROWS_, 256, 0, stream>>>(x, g1, idx_ssm, wg_ssm, cd, f_ssm, t_ssm);
    k2_wmma_zlam<<<ROWS_ / 16, 128, 0, stream>>>(t_ssm, f_ssm, U, Wgate, bgate, z, lam);
    k3a_scan_local<<<B_ * NC_, R_, 0, stream>>>(z, lam, Sloc, Pp);
    k3b_scan_chunks<<<1, B_ * R_, 0, stream>>>(Sloc, Pp, state, carry, out + (size_t)ROWS_ * D_);
    k3c_scan_fix<<<(ROWS_ * R_) / 256, 256, 0, stream>>>(Sloc, Pp, carry);
    k4_wmma_tilde_residual<<<ROWS_ / 16, 128, 0, stream>>>(Sloc, V, f_ssm, cd + P_, x, out);
    k5_mlp<<<ROWS_, 256, 0, stream>>>(out, g2, idx_mlp, wg_mlp, cd + 2 * P_, cd + 3 * P_);
}